// EquilibriumSolver_83159156785388
// MI455X (gfx1250) — compile-verified
//
#include <hip/hip_runtime.h>
#include <hip/hip_bf16.h>
#include <math.h>

// ---------------------------------------------------------------------------
// EquilibriumSolver (DEQ + Anderson) for MI455X / gfx1250.
// All GEMMs run on V_WMMA_F32_16X16X4_F32 (full fp32 precision: the 1e-4
// convergence threshold and the near-singular Anderson GTG solves cannot
// survive bf16/f16 inputs). Working set (~126 MB) is L2-resident (192 MB),
// so the problem is fp32-WMMA compute bound.
//
// GEMM: block tile 128x128, 8 waves (4 down M x 2 across N), wave tile 32x64
// (2x4 v_wmma_f32_16x16x4_f32 accumulators). Tile staging uses CDNA5 async
// global->LDS DMA (GLOBAL_LOAD_ASYNC_TO_LDS_B128, ASYNCcnt) with a
// double-buffered one-barrier-per-chunk pipeline + global_prefetch ahead.
// ---------------------------------------------------------------------------

#define B_   4
#define S_   1024
#define D_   512
#define DH_  1024
#define M_   (B_ * S_)          // 4096 rows
#define NB_  (S_ * D_)          // 524288 per-batch flat dim
#define NTOT (B_ * NB_)         // 2097152 total z elements
#define MAX_IT 10
#define MEM_   5
#define RCH    64               // reduction chunks per batch

typedef __attribute__((ext_vector_type(2))) float v2f;
typedef __attribute__((ext_vector_type(8))) float v8f;

__device__ __forceinline__ float gelu_exact(float v) {
    return 0.5f * v * (1.0f + erff(v * 0.70710678118654752440f));
}

// CDNA5 async global->LDS copy, 16B per lane, tracked by ASYNCcnt.
// vdst = per-lane LDS byte address (low 32 bits of shared pointer),
// vaddr = per-lane 64-bit global address.
__device__ __forceinline__ void async_ld16(const float* g, float* l) {
    unsigned lds = (unsigned)(size_t)l;
    asm volatile("global_load_async_to_lds_b128 %0, %1, off"
                 :: "v"(lds), "v"(g) : "memory");
}
__device__ __forceinline__ void wait_async0() {
    asm volatile("s_wait_asynccnt 0" ::: "memory");
}

// ---------------------------------------------------------------------------
// Tiled fp32 WMMA GEMM:  C(M x N) = act(A(M x K) @ W(K x N) + bias)
//   A is split across two row-major sources: columns [0,K0) from A0,
//   columns [K0,K) from A1  (implements concat([z, x]) without materializing).
//   EPI: 0 = bias only, 1 = bias + exact GELU,
//        2 = fz epilogue: C = Z + 0.5 * sigmoid(acc + bias) * Dl
// ---------------------------------------------------------------------------
#define TM 128
#define TN 128
#define TK 16

template <int EPI>
__global__ __launch_bounds__(256) void gemm_wmma(
    const float* __restrict__ A0, const float* __restrict__ A1,
    int K0, int K,
    const float* __restrict__ W, const float* __restrict__ bias,
    float* __restrict__ C, int N,
    const float* __restrict__ Zp, const float* __restrict__ Dp)
{
    __shared__ float As[2][TM][TK + 4];   // 2 x 128 x 20  (10 KB each)
    __shared__ float Ws[2][TK][TN + 4];   // 2 x 16 x 132  (8.25 KB each)

    const int tid  = threadIdx.x;
    const int lane = tid & 31;
    const int wave = tid >> 5;
    const int wm   = wave & 3;               // 4 waves down M   (32 rows each)
    const int wn   = wave >> 2;              // 2 waves across N (64 cols each)
    const int bm   = blockIdx.x * TM;
    const int bn   = blockIdx.y * TN;
    const int l15  = lane & 15;
    const int khalf = (lane >> 4) << 1;      // 0 or 2 (A/B K-pair select)

    // staging addresses (constant across chunks except the k0 term)
    const int a_row[2] = { (tid + 0)   >> 2, (tid + 256) >> 2 };   // 0..127
    const int a_kq [2] = { ((tid + 0)  & 3) << 2, ((tid + 256) & 3) << 2 };
    const int w_row[2] = { (tid + 0)   >> 5, (tid + 256) >> 5 };   // 0..15
    const int w_nq [2] = { ((tid + 0)  & 31) << 2, ((tid + 256) & 31) << 2 };
    const int K1 = K - K0;
    const int nchunks = K / TK;

    v8f acc[2][4];
#pragma unroll
    for (int i = 0; i < 2; ++i)
#pragma unroll
        for (int j = 0; j < 4; ++j)
#pragma unroll
            for (int e = 0; e < 8; ++e) acc[i][j][e] = 0.0f;

    // ---- async-stage one K-chunk into LDS buffer `buf` --------------------
    auto stage = [&](int k0, int buf) {
#pragma unroll
        for (int it = 0; it < 2; ++it) {
            int gk = k0 + a_kq[it];
            const float* asrc = (gk < K0)
                ? (A0 + (size_t)(bm + a_row[it]) * K0 + gk)
                : (A1 + (size_t)(bm + a_row[it]) * K1 + (gk - K0));
            async_ld16(asrc, &As[buf][a_row[it]][a_kq[it]]);
            async_ld16(W + (size_t)(k0 + w_row[it]) * N + bn + w_nq[it],
                       &Ws[buf][w_row[it]][w_nq[it]]);
        }
    };

    // ---- prologue: DMA chunk 0 --------------------------------------------
    stage(0, 0);
    wait_async0();
    __syncthreads();

    int cur = 0;
    for (int c = 0; c < nchunks; ++c) {
        const bool hasNext = (c + 1 < nchunks);
        // DMA chunk c+1 into the other buffer; lands while we compute.
        if (hasNext) stage((c + 1) * TK, cur ^ 1);
        if (c + 2 < nchunks) {   // global_prefetch two chunks ahead
            const int k0p = (c + 2) * TK;
            int gk = k0p + a_kq[0];
            const float* ap = (gk < K0)
                ? (A0 + (size_t)(bm + a_row[0]) * K0 + gk)
                : (A1 + (size_t)(bm + a_row[0]) * K1 + (gk - K0));
            __builtin_prefetch(ap, 0, 1);
            __builtin_prefetch(W + (size_t)(k0p + w_row[0]) * N + bn + w_nq[0], 0, 1);
        }

        // ---- compute on chunk c: 4 K-steps x 8 WMMAs ----------------------
#pragma unroll
        for (int kk = 0; kk < TK; kk += 4) {
            const int kb = kk + khalf;       // lanes 0-15: K0,K1 ; 16-31: K2,K3
            v2f afr[2], bfr[4];
#pragma unroll
            for (int i = 0; i < 2; ++i) {
                int row = wm * 32 + i * 16 + l15;        // M = lane&15
                afr[i] = *reinterpret_cast<const v2f*>(&As[cur][row][kb]);
            }
#pragma unroll
            for (int j = 0; j < 4; ++j) {
                int col = wn * 64 + j * 16 + l15;        // N = lane&15
                v2f b;
                b[0] = Ws[cur][kb][col];
                b[1] = Ws[cur][kb + 1][col];
                bfr[j] = b;
            }
#pragma unroll
            for (int i = 0; i < 2; ++i)
#pragma unroll
                for (int j = 0; j < 4; ++j)
                    acc[i][j] = __builtin_amdgcn_wmma_f32_16x16x4_f32(
                        false, afr[i], false, bfr[j],
                        (short)0, acc[i][j], false, false);
        }

        // ---- retire DMA + flip buffers (uniform predicate) ----------------
        if (hasNext) {
            wait_async0();       // own async copies landed in LDS
            __syncthreads();     // everyone's landed; cur buffer released
            cur ^= 1;
        }
    }

    // ---- epilogue: C/D layout => VGPR r holds M = r + (lane>=16 ? 8 : 0) --
    const int rbase = (lane >> 4) << 3;
#pragma unroll
    for (int i = 0; i < 2; ++i) {
#pragma unroll
        for (int j = 0; j < 4; ++j) {
            int col = bn + wn * 64 + j * 16 + l15;
            float bv = bias[col];
#pragma unroll
            for (int r = 0; r < 8; ++r) {
                int row = bm + wm * 32 + i * 16 + rbase + r;
                size_t off = (size_t)row * N + col;
                float v = acc[i][j][r] + bv;
                if (EPI == 1) v = gelu_exact(v);
                if (EPI == 2) {
                    float g = 1.0f / (1.0f + expf(-v));
                    v = Zp[off] + 0.5f * g * Dp[off];
                }
                C[off] = v;
            }
        }
    }
}

// ---------------------------------------------------------------------------
// Scalar scratch layout (floats) at `scal`:
//   [0..99]    GTG[b*25 + p*5 + q] (p<=q)     [100..119] alpha[b*5+e]
//   [120] residual  [121] res_fin  [122] conv  [123] just  [124] it_fin
// ---------------------------------------------------------------------------
#define SC_GTG  0
#define SC_ALP  100
#define SC_RES  120
#define SC_RFIN 121
#define SC_CONV 122
#define SC_JUST 123
#define SC_IT   124

__global__ void init_scal_kernel(float* scal) {
    int t = threadIdx.x;
    if (t < 125) scal[t] = 0.0f;
    if (t == 0) scal[SC_IT] = (float)MAX_IT;
}

__global__ void zero_gtg_kernel(float* scal) {
    int t = threadIdx.x;
    if (t < 100) scal[SC_GTG + t] = 0.0f;
}

// Gram matrix of residuals: GTG[b][p][q] = <F_p - X_p, F_q - X_q>  (p<=q<k).
// One pass over the data computes all 15 upper-triangular entries.
__global__ __launch_bounds__(256) void gtg_kernel(
    const float* __restrict__ Xb, const float* __restrict__ Fb,
    float* __restrict__ scal, int iter, int k)
{
    const int PP[15] = {0,0,0,0,0,1,1,1,1,2,2,2,3,3,4};
    const int QQ[15] = {0,1,2,3,4,1,2,3,4,2,3,4,3,4,4};

    int b  = blockIdx.x / RCH;
    int ch = blockIdx.x % RCH;
    int slot[MEM_];
#pragma unroll
    for (int e = 0; e < MEM_; ++e) slot[e] = (iter - (k - 1) + e) % MEM_;

    size_t base = (size_t)b * NB_ + (size_t)ch * (NB_ / RCH);
    float accv[15];
#pragma unroll
    for (int t = 0; t < 15; ++t) accv[t] = 0.0f;

    for (int idx = threadIdx.x; idx < NB_ / RCH; idx += 256) {
        size_t o = base + idx;
        float g[MEM_];
#pragma unroll
        for (int e = 0; e < MEM_; ++e) {
            g[e] = (e < k) ? (Fb[(size_t)slot[e] * NTOT + o] -
                              Xb[(size_t)slot[e] * NTOT + o])
                           : 0.0f;
        }
#pragma unroll
        for (int t = 0; t < 15; ++t) accv[t] += g[PP[t]] * g[QQ[t]];
    }
#pragma unroll
    for (int t = 0; t < 15; ++t) {
        float v = accv[t];
        for (int off = 16; off > 0; off >>= 1) v += __shfl_xor(v, off, 32);
        if ((threadIdx.x & 31) == 0 && PP[t] < k && QQ[t] < k)
            atomicAdd(&scal[SC_GTG + b * 25 + PP[t] * 5 + QQ[t]], v);
    }
}

// residual = mean_b sqrt(GTG diag of newest entry); freeze-on-converge latch.
__global__ void update_scalars_kernel(float* scal, int iter, int k) {
    if (threadIdx.x != 0 || blockIdx.x != 0) return;
    int d = (k - 1) * 5 + (k - 1);
    float s = 0.0f;
    for (int b = 0; b < B_; ++b) s += sqrtf(scal[SC_GTG + b * 25 + d]);
    float resid = s * (1.0f / (float)B_);
    scal[SC_RES] = resid;
    float conv = scal[SC_CONV];
    float just = (conv == 0.0f && resid < 1e-4f) ? 1.0f : 0.0f;
    scal[SC_JUST] = just;
    if (just != 0.0f) {
        scal[SC_RFIN] = resid;
        scal[SC_IT]   = (float)(iter + 1);
        scal[SC_CONV] = 1.0f;
    }
}

// Per-batch (GTG + 1e-6 I) alpha = 1, pivoted Gaussian elimination, normalize.
__global__ void solve_kernel(float* scal, int k) {
    int b = threadIdx.x;
    if (b >= B_ || blockIdx.x != 0) return;
    float Mm[MEM_][MEM_], rhs[MEM_];
    for (int p = 0; p < k; ++p) {
        for (int q = 0; q < k; ++q) {
            int pp = p < q ? p : q, qq = p < q ? q : p;
            Mm[p][q] = scal[SC_GTG + b * 25 + pp * 5 + qq];
        }
        Mm[p][p] += 1e-6f;
        rhs[p] = 1.0f;
    }
    for (int c = 0; c < k; ++c) {
        int piv = c; float mx = fabsf(Mm[c][c]);
        for (int r = c + 1; r < k; ++r) {
            float a = fabsf(Mm[r][c]);
            if (a > mx) { mx = a; piv = r; }
        }
        if (piv != c) {
            for (int q = c; q < k; ++q) { float t = Mm[c][q]; Mm[c][q] = Mm[piv][q]; Mm[piv][q] = t; }
            float t = rhs[c]; rhs[c] = rhs[piv]; rhs[piv] = t;
        }
        float inv = 1.0f / Mm[c][c];
        for (int r = c + 1; r < k; ++r) {
            float f = Mm[r][c] * inv;
            for (int q = c; q < k; ++q) Mm[r][q] -= f * Mm[c][q];
            rhs[r] -= f * rhs[c];
        }
    }
    for (int c = k - 1; c >= 0; --c) {
        float v = rhs[c];
        for (int q = c + 1; q < k; ++q) v -= Mm[c][q] * rhs[q];
        rhs[c] = v / Mm[c][c];
    }
    float s = 0.0f;
    for (int e = 0; e < k; ++e) s += rhs[e];
    float invs = 1.0f / s;
    for (int e = 0; e < MEM_; ++e)
        scal[SC_ALP + b * 5 + e] = (e < k) ? rhs[e] * invs : 0.0f;
}

// zf_{i+1} = sum_e alpha[b][e] * F[slot_e]   (k==1 reduces to zf=fz exactly)
__global__ __launch_bounds__(256) void combine_kernel(
    const float* __restrict__ Fb, const float* __restrict__ scal,
    float* __restrict__ zNew, int iter, int k)
{
    int idx = blockIdx.x * 256 + threadIdx.x;
    if (idx >= NTOT) return;
    int b = idx / NB_;
    float v = 0.0f;
    for (int e = 0; e < k; ++e) {
        int slot = (iter - (k - 1) + e) % MEM_;
        v += scal[SC_ALP + b * 5 + e] * Fb[(size_t)slot * NTOT + idx];
    }
    zNew[idx] = v;
}

// Latch z_out = fz on the just-converged iteration (uniform predicate).
__global__ __launch_bounds__(256) void freeze_kernel(
    const float* __restrict__ Fs, const float* __restrict__ scal,
    float* __restrict__ zout)
{
    if (scal[SC_JUST] == 0.0f) return;
    int idx = blockIdx.x * 256 + threadIdx.x;
    if (idx < NTOT) zout[idx] = Fs[idx];
}

// Final select + scalar tail:  (z_out | zf_final), it_fin, conv, res_fin.
__global__ __launch_bounds__(256) void final_kernel(
    const float* __restrict__ Xfin, const float* __restrict__ zout,
    const float* __restrict__ scal, float* __restrict__ out, int out_size)
{
    int idx = blockIdx.x * 256 + threadIdx.x;
    bool conv = (scal[SC_CONV] != 0.0f);
    if (idx < NTOT) out[idx] = conv ? zout[idx] : Xfin[idx];
    if (idx == 0) {
        if (NTOT + 0 < out_size) out[NTOT + 0] = scal[SC_IT];
        if (NTOT + 1 < out_size) out[NTOT + 1] = conv ? 1.0f : 0.0f;
        if (NTOT + 2 < out_size) out[NTOT + 2] = conv ? scal[SC_RFIN] : scal[SC_RES];
    }
}

// ---------------------------------------------------------------------------
// Host orchestration. Workspace layout (floats):
//   X ring  [5*NTOT] | F ring [5*NTOT] | h1 [2*NTOT] | h2 [2*NTOT]
//   zout [NTOT] | scal [128]       => ~126 MB total, all L2-resident.
// ---------------------------------------------------------------------------
extern "C" void kernel_launch(void* const* d_in, const int* in_sizes, int n_in,
                              void* d_out, int out_size, void* d_ws, size_t ws_size,
                              hipStream_t stream) {
    const float* x   = (const float*)d_in[0];
    const float* Wi1 = (const float*)d_in[1];
    const float* bi1 = (const float*)d_in[2];
    const float* Wi2 = (const float*)d_in[3];
    const float* bi2 = (const float*)d_in[4];
    const float* Wt1 = (const float*)d_in[5];
    const float* bt1 = (const float*)d_in[6];
    const float* Wt2 = (const float*)d_in[7];
    const float* bt2 = (const float*)d_in[8];
    const float* Wt3 = (const float*)d_in[9];
    const float* bt3 = (const float*)d_in[10];
    const float* Wg  = (const float*)d_in[11];
    const float* bg  = (const float*)d_in[12];

    float* ws   = (float*)d_ws;
    float* X    = ws;                                // 5 ring slots
    float* F    = ws + 5ull * NTOT;                  // 5 ring slots
    float* h1   = ws + 10ull * NTOT;                 // 4096x1024
    float* h2   = ws + 12ull * NTOT;                 // 4096x1024
    float* zout = ws + 14ull * NTOT;
    float* scal = ws + 15ull * NTOT;

    const dim3 blk(256);
    const dim3 gN1024(M_ / TM, DH_ / TN);            // 32 x 8
    const dim3 gN512 (M_ / TM, D_  / TN);            // 32 x 4
    const int  elemBlocks = (NTOT + 255) / 256;

    init_scal_kernel<<<1, 128, 0, stream>>>(scal);

    // z0 = gelu(x @ Wi1 + bi1) @ Wi2 + bi2  -> X[0]
    gemm_wmma<1><<<gN1024, blk, 0, stream>>>(x,  x,  D_,  D_,  Wi1, bi1, h1, DH_, nullptr, nullptr);
    gemm_wmma<0><<<gN512,  blk, 0, stream>>>(h1, h1, DH_, DH_, Wi2, bi2, X,  D_,  nullptr, nullptr);

    for (int i = 0; i < MAX_IT; ++i) {
        const int s = i % MEM_;
        const int k = (i + 1 < MEM_) ? (i + 1) : MEM_;
        float* Xs = X + (size_t)s * NTOT;            // zf of this iteration
        float* Fs = F + (size_t)s * NTOT;            // fz of this iteration
        float* Xn = X + (size_t)((i + 1) % MEM_) * NTOT;

        zero_gtg_kernel<<<1, 128, 0, stream>>>(scal);

        // h1 = gelu([zf, x] @ Wt1 + bt1)        (concat via split-K sources)
        gemm_wmma<1><<<gN1024, blk, 0, stream>>>(Xs, x,  D_,  2 * D_, Wt1, bt1, h1, DH_, nullptr, nullptr);
        // h2 = gelu(h1 @ Wt2 + bt2)
        gemm_wmma<1><<<gN1024, blk, 0, stream>>>(h1, h1, DH_, DH_,    Wt2, bt2, h2, DH_, nullptr, nullptr);
        // delta = h2 @ Wt3 + bt3   (h1 is free now -> reuse as delta buffer)
        gemm_wmma<0><<<gN512,  blk, 0, stream>>>(h2, h2, DH_, DH_,    Wt3, bt3, h1, D_,  nullptr, nullptr);
        // fz = zf + 0.5 * sigmoid([zf,x] @ Wg + bg) * delta   -> F[s]
        gemm_wmma<2><<<gN512,  blk, 0, stream>>>(Xs, x,  D_,  2 * D_, Wg,  bg,  Fs, D_,  Xs, h1);

        gtg_kernel<<<B_ * RCH, blk, 0, stream>>>(X, F, scal, i, k);
        update_scalars_kernel<<<1, 32, 0, stream>>>(scal, i, k);
        solve_kernel<<<1, 32, 0, stream>>>(scal, k);
        freeze_kernel<<<elemBlocks, blk, 0, stream>>>(Fs, scal, zout);
        combine_kernel<<<elemBlocks, blk, 0, stream>>>(F, scal, Xn, i, k);
    }

    // zf after loop sits in X[MAX_IT % 5] == X[0]
    final_kernel<<<elemBlocks, blk, 0, stream>>>(
        X + (size_t)(MAX_IT % MEM_) * NTOT, zout, scal, (float*)d_out, out_size);
}